// OctahedralSHT_335007449424
// MI455X (gfx1250) — compile-verified
//
#include <hip/hip_runtime.h>

typedef __attribute__((ext_vector_type(2))) float v2f;
typedef __attribute__((ext_vector_type(8))) float v8f;

#define NLAT   192
#define MMAX   96
#define LMAX   96
#define BROWS  400              // b*t*e*v = 4*100
#define NPTS   40320
#define MAXN   400
// packed dft:  pk[c][j][n/2][m]  (v2f) : 2*192*200*96 = 7,372,800 v2f = 59 MB
#define PK_V2F      7372800L
// f_hat:       fh[c][m][j/2][row] (v2f): 2*96*96*400  = 7,372,800 v2f = 59 MB
#define FHAT_PLANE_V2F 3686400L  // 96*96*400

// ---------------------------------------------------------------------------
// K0: pack dft (complex64, [192][400][96]) into planar re/im, K-pair layout:
//   pk[((c*192 + j)*200 + p)*96 + m] = { dft[j][2p][m].c , dft[j][2p+1][m].c }
// grid = (75, 384=c*192+j), block = 256   (75*256 == 200*96 exactly)
// ---------------------------------------------------------------------------
__global__ void sht_pack_dft(const float* __restrict__ dftf,
                             v2f* __restrict__ pk) {
    int t  = blockIdx.x * 256 + threadIdx.x;   // 0..19199
    int m  = t % 96;
    int p  = t / 96;                           // n-pair index 0..199
    int cj = blockIdx.y;                       // c*192 + j
    int c  = cj / 192;
    int j  = cj - c * 192;
    long s = (((long)j * 400 + 2 * p) * 96 + m) * 2 + c;  // float idx of dft[j][2p][m].c
    v2f o;
    o.x = dftf[s];
    o.y = dftf[s + 192];                       // next n: +96 complex = +192 floats
    pk[((long)cj * 200 + p) * 96 + m] = o;
}

// ---------------------------------------------------------------------------
// K1: per-ring truncated real DFT via V_WMMA_F32_16X16X4_F32.
//   fh[c][m][j/2][row].(x|y) = sum_n data-row * dft, ragged ring lengths.
// grid = (192 rings, 25 row-tiles), block = 1 wave (32 lanes)
// ---------------------------------------------------------------------------
__global__ __launch_bounds__(32) void sht_dft_stage(const float* __restrict__ data,
                                                    const v2f* __restrict__ pk,
                                                    float* __restrict__ fhatf) {
    const int j    = blockIdx.x;
    const int rt   = blockIdx.y;
    const int lane = threadIdx.x;
    const int mloc = lane & 15;
    const int hf   = lane >> 4;        // 0 | 1
    const int koff = hf << 1;          // K offset 0 | 2 (ISA 32-bit A/B layout)

    int nlen, nstart;
    if (j < 96) { nlen = 20 + 4 * j;  nstart = 2 * j * j + 18 * j; }
    else        { int h = 191 - j; nlen = 20 + 4 * h;
                  nstart = NPTS - (h + 1) * (2 * h + 20); }

    v8f acc[12];
    #pragma unroll
    for (int i = 0; i < 12; ++i) acc[i] = (v8f){0,0,0,0,0,0,0,0};

    // A fragment source: direct ragged read of data[b][point][v], row = b*100+v.
    const int row = rt * 16 + mloc;
    const int b   = row / 100;
    const int v   = row - b * 100;
    const float* abase = data + ((long)b * NPTS + nstart + koff) * 100 + v;

    // B fragment sources: packed planar re/im, single b64 load each.
    const v2f* pkre = pk + ((long)j * 200 + hf) * 96;            // c = 0
    const v2f* pkim = pkre + (long)192 * 200 * 96;               // c = 1

    for (int k = 0; k < nlen; k += 4) {
        v2f a;
        a.x = abase[(long)k * 100];        // x[row][nstart+k+koff]
        a.y = abase[(long)k * 100 + 100];  // x[row][nstart+k+koff+1]
        long kofs = (long)(k >> 1) * 96;
        #pragma unroll
        for (int mt = 0; mt < 6; ++mt) {
            int m = mt * 16 + mloc;                 // N index of this tile
            v2f bre = pkre[kofs + m];
            v2f bim = pkim[kofs + m];
            acc[2 * mt] = __builtin_amdgcn_wmma_f32_16x16x4_f32(
                false, a, false, bre, (short)0, acc[2 * mt], false, false);
            acc[2 * mt + 1] = __builtin_amdgcn_wmma_f32_16x16x4_f32(
                false, a, false, bim, (short)0, acc[2 * mt + 1], false, false);
        }
    }

    // Epilogue: LDS-transpose C frags; store into j-pair-interleaved f_hat.
    __shared__ float tile[16][17];
    const int jp = j >> 1, jc = j & 1;
    for (int q = 0; q < 12; ++q) {
        int mt = q >> 1, c = q & 1;
        __syncthreads();
        #pragma unroll
        for (int i = 0; i < 8; ++i)
            tile[mloc][i + hf * 8] = acc[q][i];   // tile[N=m_local][M=row_local]
        __syncthreads();
        for (int t = lane; t < 256; t += 32) {
            int Nn = t >> 4, Mm = t & 15;
            int m  = mt * 16 + Nn;
            long idx = ((((long)c * MMAX + m) * 96 + jp) * BROWS + rt * 16 + Mm) * 2 + jc;
            fhatf[idx] = tile[Nn][Mm];
        }
    }
}

// ---------------------------------------------------------------------------
// K2: Legendre contraction per m via V_WMMA_F32_16X16X4_F32.
//   coeffs[row][l][m].{re,im} = sum_j fh[c][m][j][row] * leg[l][m][j]
// grid = (96 m-values, 25 row-tiles), block = 1 wave
// ---------------------------------------------------------------------------
__global__ __launch_bounds__(32) void sht_leg_stage(const float* __restrict__ fhatf,
                                                    const float* __restrict__ leg,
                                                    float* __restrict__ out) {
    const int m    = blockIdx.x;
    const int rt   = blockIdx.y;
    const int lane = threadIdx.x;
    const int mloc = lane & 15;
    const int hf   = lane >> 4;
    const int koff = hf << 1;

    v8f accre[6], accim[6];
    #pragma unroll
    for (int i = 0; i < 6; ++i) {
        accre[i] = (v8f){0,0,0,0,0,0,0,0};
        accim[i] = (v8f){0,0,0,0,0,0,0,0};
    }

    const int row = rt * 16 + mloc;
    // A fragments: one b64 load per plane from j-pair-interleaved f_hat.
    const v2f* fh2     = (const v2f*)fhatf;
    const v2f* arebase = fh2 + ((long)m * 96 + hf) * BROWS + row;       // c = 0
    const v2f* aimbase = arebase + FHAT_PLANE_V2F;                      // c = 1

    for (int j = 0; j < NLAT; j += 4) {
        long aofs = (long)(j >> 1) * BROWS;
        v2f are = arebase[aofs];    // {fh[m][j+koff][row], fh[m][j+koff+1][row]}
        v2f aim = aimbase[aofs];
        long lb = (long)m * NLAT + j + koff;   // leg idx = (l*96 + m)*192 + n
        #pragma unroll
        for (int lt = 0; lt < 6; ++lt) {
            int l = lt * 16 + mloc;
            v2f bb = *(const v2f*)(leg + (long)l * (MMAX * NLAT) + lb);
            accre[lt] = __builtin_amdgcn_wmma_f32_16x16x4_f32(
                false, are, false, bb, (short)0, accre[lt], false, false);
            accim[lt] = __builtin_amdgcn_wmma_f32_16x16x4_f32(
                false, aim, false, bb, (short)0, accim[lt], false, false);
        }
    }

    // Epilogue: LDS transpose, then interleaved complex float2 stores in
    // out[b][l][m][v][2] layout (row = b*100 + v).
    __shared__ float tre[16][17];
    __shared__ float tim[16][17];
    for (int lt = 0; lt < 6; ++lt) {
        __syncthreads();
        #pragma unroll
        for (int i = 0; i < 8; ++i) {
            tre[mloc][i + hf * 8] = accre[lt][i];
            tim[mloc][i + hf * 8] = accim[lt][i];
        }
        __syncthreads();
        for (int t = lane; t < 256; t += 32) {
            int Nn = t >> 4, Mm = t & 15;
            int r  = rt * 16 + Mm;
            int bb = r / 100;
            int vv = r - bb * 100;
            int l  = lt * 16 + Nn;
            long idx = ((((long)bb * LMAX + l) * MMAX + m) * 100 + vv) * 2;
            v2f val = {tre[Nn][Mm], tim[Nn][Mm]};
            *(v2f*)(out + idx) = val;
        }
    }
}

// ---------------------------------------------------------------------------
extern "C" void kernel_launch(void* const* d_in, const int* in_sizes, int n_in,
                              void* d_out, int out_size, void* d_ws, size_t ws_size,
                              hipStream_t stream) {
    const float* data = (const float*)d_in[0];   // [4][40320][100] f32
    const float* dftf = (const float*)d_in[1];   // [192][400][96] complex64 -> f32 pairs
    const float* legw = (const float*)d_in[2];   // [96][96][192] f32
    float* out = (float*)d_out;                  // [4][96][96][100] complex64 -> f32 pairs

    float* wsf   = (float*)d_ws;
    v2f*   pk    = (v2f*)wsf;                    // 7,372,800 v2f (59 MB)
    float* fhatf = wsf + 2 * PK_V2F;             // 14,745,600 f32 (59 MB)

    sht_pack_dft <<<dim3(75, 384),       256, 0, stream>>>(dftf, pk);
    sht_dft_stage<<<dim3(NLAT, BROWS/16), 32, 0, stream>>>(data, pk, fhatf);
    sht_leg_stage<<<dim3(MMAX, BROWS/16), 32, 0, stream>>>(fhatf, legw, out);
}